// EnhancedUberCRSN_24902220382545
// MI455X (gfx1250) — compile-verified
//
#include <hip/hip_runtime.h>
#include <math.h>

// ---------- CDNA5 wave32 WMMA fp32 path ----------
typedef float v2f __attribute__((ext_vector_type(2)));
typedef float v8f __attribute__((ext_vector_type(8)));

#define NB 256      // batch
#define ND 256      // D
#define NS 32       // stack slots
#define NF 512      // 2*D
#define NK 512      // codebook entries
#define NDEPTH 8
#define C_EPS 1e-6f
#define C_LAM 0.1f
#define C_RESW 0.1f
#define ATTN_SCALE 0.0625f   // 256^-0.5
#define KQTOT 128            // 512/4 k-quads

__device__ __forceinline__ v8f wmma4(v2f a, v2f b, v8f c) {
  // V_WMMA_F32_16X16X4_F32 : D(16x16 f32) = A(16x4) * B(4x16) + C
  return __builtin_amdgcn_wmma_f32_16x16x4_f32(false, a, false, b, (short)0, c,
                                               false, false);
}

__device__ __forceinline__ float sigm(float x) { return 1.0f / (1.0f + expf(-x)); }

__device__ __forceinline__ float block_reduce_sum(float v, float* buf) {
  int tid = threadIdx.x;
  buf[tid] = v;
  __syncthreads();
  for (int s = 128; s > 0; s >>= 1) {
    if (tid < s) buf[tid] += buf[tid + s];
    __syncthreads();
  }
  float r = buf[0];
  __syncthreads();
  return r;
}

// ---------------- init ----------------
__global__ void k_init(const float* __restrict__ zr, const float* __restrict__ zi,
                       const float* __restrict__ mem0, const float* __restrict__ ptr0,
                       float* zf, float* mem, float* ptr, float* rem, float* probs,
                       float* acc) {
  int idx = blockIdx.x * 256 + threadIdx.x;   // grid covers NB*NS*NF exactly
  mem[idx] = mem0[idx];
  if (idx < NB * NF) {
    int b = idx >> 9, j = idx & (NF - 1);
    zf[idx] = (j < ND) ? zr[b * ND + j] : zi[b * ND + (j - ND)];
    acc[idx] = 0.0f;
    probs[idx] = 0.0f;   // NB*NK == NB*NF
  }
  if (idx < NB * NS) ptr[idx] = ptr0[idx];
  if (idx < NB) rem[idx] = 1.0f;
}

// -------- pack effective complex weight into WMMA B-fragment tile order -----
// Bpack[((ntile*KQTOT + kq)*32 + lane)] holds float2:
//   col = ntile*16 + (lane&15), r = lane>>4, k0 = 4*kq + 2*r  -> (B[k0][col], B[k0+1][col])
__global__ void k_pack_weff(const float* __restrict__ Wr, const float* __restrict__ Wi,
                            float* __restrict__ pack) {
  int t = blockIdx.x * 256 + threadIdx.x;   // 131072 threads total
  int lane = t & 31, rest = t >> 5;
  int kq = rest & (KQTOT - 1), ntile = rest >> 7;
  int col = ntile * 16 + (lane & 15);
  int k0 = kq * 4 + 2 * (lane >> 4);
#pragma unroll
  for (int ii = 0; ii < 2; ii++) {
    int kk = k0 + ii;
    float val;
    if (col < ND)
      val = (kk < ND) ? Wr[col * ND + kk] : -Wi[col * ND + (kk - ND)];
    else
      val = (kk < ND) ? Wi[(col - ND) * ND + kk] : Wr[(col - ND) * ND + (kk - ND)];
    pack[t * 2 + ii] = val;
  }
}

__global__ void k_pack_mat(const float* __restrict__ src, float* __restrict__ pack,
                           int transpose) {
  int t = blockIdx.x * 256 + threadIdx.x;
  int lane = t & 31, rest = t >> 5;
  int kq = rest & (KQTOT - 1), ntile = rest >> 7;
  int col = ntile * 16 + (lane & 15);
  int k0 = kq * 4 + 2 * (lane >> 4);
#pragma unroll
  for (int ii = 0; ii < 2; ii++) {
    int kk = k0 + ii;
    pack[t * 2 + ii] = transpose ? src[col * NF + kk] : src[kk * NF + col];
  }
}

__global__ void k_cbsq(const float* __restrict__ cb, float* __restrict__ cbsq) {
  int j = blockIdx.x * 256 + threadIdx.x;
  if (j < NK) {
    float s = 0.0f;
    for (int d = 0; d < NF; d++) { float c = cb[j * NF + d]; s += c * c; }
    cbsq[j] = s;
  }
}

// ------------- gates + ptr roll + mem blend (one block per batch) -----------
__global__ void k_ctrl(const float* __restrict__ Wc, const float* __restrict__ bc,
                       float* __restrict__ zf, float* __restrict__ ptr,
                       float* __restrict__ mem) {
  __shared__ float zfl[NF];
  __shared__ float pl[NS];
  __shared__ float buf[256];
  __shared__ float gsh[3];
  __shared__ float pps[3];
  int b = blockIdx.x, tid = threadIdx.x;
  zfl[tid] = zf[b * NF + tid];
  zfl[tid + 256] = zf[b * NF + 256 + tid];
  if (tid < NS) pl[tid] = ptr[b * NS + tid];
  __syncthreads();
  float p0 = 0, p1 = 0, p2 = 0;
  for (int j = tid; j < NF; j += 256) {
    float z = zfl[j];
    p0 += z * Wc[j * 3 + 0];
    p1 += z * Wc[j * 3 + 1];
    p2 += z * Wc[j * 3 + 2];
  }
  float s;
  s = block_reduce_sum(p0, buf); if (tid == 0) gsh[0] = sigm(s + bc[0]);
  s = block_reduce_sum(p1, buf); if (tid == 0) gsh[1] = sigm(s + bc[1]);
  s = block_reduce_sum(p2, buf); if (tid == 0) gsh[2] = sigm(s + bc[2]);
  __syncthreads();
  if (tid == 0) {
    float tot = gsh[0] + gsh[1] + gsh[2] + C_EPS;
    pps[0] = gsh[0] / tot; pps[1] = gsh[1] / tot; pps[2] = gsh[2] / tot;
  }
  __syncthreads();
  float push = pps[0], pop = pps[1], stay = pps[2];
  if (tid < NS) {
    float np = push * pl[(tid + NS - 1) & 31] + pop * pl[(tid + 1) & 31] + stay * pl[tid];
    ptr[b * NS + tid] = np;
  }
  float om = 1.0f - push;
  size_t base = (size_t)b * NS * NF;
  for (int i = tid; i < NS * NF; i += 256)
    mem[base + i] = mem[base + i] * om + push * zfl[i & (NF - 1)];
}

// -------- fused per-batch: Q/K GEMMs (WMMA), scores (WMMA), softmax, pvec, mbar
__global__ void k_attn(const float* __restrict__ mem, const float* __restrict__ Wq_pack,
                       const float* __restrict__ Wk_pack, const float* __restrict__ ptr,
                       float* __restrict__ mbar) {
  extern __shared__ float sm[];
  float* mems = sm;                 // 32*512
  float* qs   = sm + NS * NF;       // 32*512
  float* ks   = sm + 2 * NS * NF;   // 32*512
  float* sc   = sm + 3 * NS * NF;   // 32*32
  float* ptrl = sc + NS * NS;       // 32
  float* pv   = ptrl + NS;          // 32
  int b = blockIdx.x, tid = threadIdx.x;
  int w = tid >> 5, l = tid & 31;

  // cooperative load of mem[b] (32x512)
  const float4* msrc = (const float4*)(mem + (size_t)b * NS * NF);
  float4* mdst = (float4*)mems;
#pragma unroll
  for (int i = 0; i < 16; i++) mdst[tid + 256 * i] = msrc[tid + 256 * i];
  if (tid < NS) ptrl[tid] = ptr[b * NS + tid];
  __syncthreads();

  // Q = mems @ Wq_eff ; K = mems @ Wk_eff  (M=32,N=512,K=512)  -- all WMMA
  for (int which = 0; which < 2; which++) {
    const v2f* bp = (const v2f*)(which ? Wk_pack : Wq_pack);
    float* dst = which ? ks : qs;
    for (int i = 0; i < 8; i++) {
      int tau = w * 8 + i;                 // 64 tiles of 16x16
      int rt = tau & 1, ct = tau >> 1;
      int abase = (16 * rt + (l & 15)) * NF + 2 * (l >> 4);
      const v2f* bcol = bp + (size_t)ct * KQTOT * 32 + l;
      v8f acc = {};
      for (int kq = 0; kq < KQTOT; kq++) {
        v2f a = *(const v2f*)&mems[abase + 4 * kq];
        v2f bb = bcol[kq * 32];
        acc = wmma4(a, bb, acc);
      }
      int crow0 = 16 * rt + 8 * (l >> 4);
      int ccol = ct * 16 + (l & 15);
#pragma unroll
      for (int v = 0; v < 8; v++) dst[(crow0 + v) * NF + ccol] = acc[v];
    }
  }
  __syncthreads();

  // scores (32x32) = Q @ K^T * scale  -- waves 0..3, one 16x16 tile each
  if (w < 4) {
    int sr = (w & 1) * 16, tc = (w >> 1) * 16;
    int abase = (sr + (l & 15)) * NF + 2 * (l >> 4);
    int bbase = (tc + (l & 15)) * NF + 2 * (l >> 4);
    v8f acc = {};
    for (int kq = 0; kq < KQTOT; kq++) {
      v2f a = *(const v2f*)&qs[abase + 4 * kq];
      v2f bb = *(const v2f*)&ks[bbase + 4 * kq];
      acc = wmma4(a, bb, acc);
    }
    int crow0 = sr + 8 * (l >> 4);
    int ccol = tc + (l & 15);
#pragma unroll
    for (int v = 0; v < 8; v++) sc[(crow0 + v) * NS + ccol] = acc[v] * ATTN_SCALE;
  }
  __syncthreads();

  // row softmax
  if (tid < NS) {
    float m = -1e30f;
    for (int t2 = 0; t2 < NS; t2++) m = fmaxf(m, sc[tid * NS + t2]);
    float ssum = 0.0f;
    for (int t2 = 0; t2 < NS; t2++) {
      float e = expf(sc[tid * NS + t2] - m);
      sc[tid * NS + t2] = e;
      ssum += e;
    }
    float inv = 1.0f / ssum;
    for (int t2 = 0; t2 < NS; t2++) sc[tid * NS + t2] *= inv;
  }
  __syncthreads();
  // pvec = w^T @ new_ptr
  if (tid < NS) {
    float a2 = 0.0f;
    for (int s2 = 0; s2 < NS; s2++) a2 += ptrl[s2] * sc[s2 * NS + tid];
    pv[tid] = a2;
  }
  __syncthreads();
  // mbar = pvec @ mem[b]   (exact collapse of read = (ptr^T w) @ V(mem))
  for (int j = tid; j < NF; j += 256) {
    float a3 = 0.0f;
    for (int t2 = 0; t2 < NS; t2++) a3 += pv[t2] * mems[t2 * NF + j];
    mbar[b * NF + j] = a3;
  }
}

// -------- generic WMMA GEMM: C[Mx512] = A[Mx512] @ Bpack, 64x64 block tile ---
__global__ void k_gemm(const float* __restrict__ A, const float* __restrict__ Bpack,
                       float* __restrict__ C) {
  __shared__ float Atile[64 * 64];
  int m0 = blockIdx.x * 64, n0 = blockIdx.y * 64;
  int tid = threadIdx.x, w = tid >> 5, l = tid & 31;
  int rt = w & 3;          // 16-row slice
  int cp = w >> 2;         // column pair
  int ntile0 = (n0 >> 4) + cp * 2;
  const v2f* bp = (const v2f*)Bpack;
  v8f acc0 = {}, acc1 = {};
  for (int kb = 0; kb < 8; kb++) {
#pragma unroll
    for (int i = 0; i < 4; i++) {
      int idx4 = tid + 256 * i;        // 1024 float4s
      int row = idx4 >> 4, c4 = idx4 & 15;
      *(float4*)&Atile[row * 64 + c4 * 4] =
          *(const float4*)&A[(size_t)(m0 + row) * NF + kb * 64 + c4 * 4];
    }
    __syncthreads();
    int abase = (16 * rt + (l & 15)) * 64 + 2 * (l >> 4);
#pragma unroll
    for (int kq = 0; kq < 16; kq++) {
      int gk = kb * 16 + kq;
      v2f a = *(const v2f*)&Atile[abase + 4 * kq];
      v2f b0 = bp[((size_t)ntile0 * KQTOT + gk) * 32 + l];
      v2f b1 = bp[((size_t)(ntile0 + 1) * KQTOT + gk) * 32 + l];
      acc0 = wmma4(a, b0, acc0);
      acc1 = wmma4(a, b1, acc1);
    }
    __syncthreads();
  }
  int row0 = m0 + 16 * rt + 8 * (l >> 4);
  int col = ntile0 * 16 + (l & 15);
#pragma unroll
  for (int v = 0; v < 8; v++) {
    C[(size_t)(row0 + v) * NF + col] = acc0[v];
    C[(size_t)(row0 + v) * NF + col + 16] = acc1[v];
  }
}

// -------- zf2 = zf + 0.1*Rv ; per-batch mag/znorm partials ------------------
__global__ void k_post(const float* __restrict__ zf, const float* __restrict__ Rv,
                       float* __restrict__ zf2, float* __restrict__ magsum,
                       float* __restrict__ magsq, float* __restrict__ znorm) {
  __shared__ float buf[256];
  int b = blockIdx.x, tid = threadIdx.x;
  float a0 = zf[b * NF + tid] + C_RESW * Rv[b * NF + tid];
  float a1 = zf[b * NF + 256 + tid] + C_RESW * Rv[b * NF + 256 + tid];
  zf2[b * NF + tid] = a0;
  zf2[b * NF + 256 + tid] = a1;
  float nrm2 = a0 * a0 + a1 * a1;
  float mg = sqrtf(nrm2 + C_EPS);
  float s;
  s = block_reduce_sum(mg, buf);      if (tid == 0) magsum[b] = s;
  s = block_reduce_sum(mg * mg, buf); if (tid == 0) magsq[b] = s;
  s = block_reduce_sum(nrm2, buf);    if (tid == 0) znorm[b] = s;
}

__global__ void k_var(const float* __restrict__ magsum, const float* __restrict__ magsq,
                      float* __restrict__ scal) {
  __shared__ float buf[256];
  int tid = threadIdx.x;
  float s1 = block_reduce_sum(magsum[tid], buf);
  float s2 = block_reduce_sum(magsq[tid], buf);
  if (tid == 0) {
    float n = (float)(NB * ND);
    float mean = s1 / n;
    float var = s2 / n - mean * mean;
    float x = var / (1.0f + C_EPS);
    scal[0] = (x > 20.0f) ? x : log1pf(expf(x));   // softplus
  }
}

// -------- distances + adjacency bias + softmax -> probs ---------------------
__global__ void k_dist(const float* __restrict__ G1, const float* __restrict__ G2,
                       const float* __restrict__ cbsq, const float* __restrict__ znorm,
                       const float* __restrict__ scal, float* __restrict__ probs,
                       int step) {
  __shared__ float dl[NK];
  __shared__ float buf[256];
  int b = blockIdx.x, tid = threadIdx.x;
  float mod = (step > 0) ? (C_LAM * scal[0]) : 0.0f;
  float zn = znorm[b];
  for (int j = tid; j < NK; j += 256) {
    float d = (zn + cbsq[j] - 2.0f * G1[b * NK + j]) * (1.0f / (float)NF);
    float bias = sigm(G2[b * NK + j]);
    dl[j] = -(d - mod * bias);    // logits, TEMP=1
  }
  __syncthreads();
  float m = fmaxf(dl[tid], dl[tid + 256]);
  buf[tid] = m;
  __syncthreads();
  for (int s = 128; s > 0; s >>= 1) {
    if (tid < s) buf[tid] = fmaxf(buf[tid], buf[tid + s]);
    __syncthreads();
  }
  m = buf[0];
  __syncthreads();
  float e0 = expf(dl[tid] - m), e1 = expf(dl[tid + 256] - m);
  float ssum = block_reduce_sum(e0 + e1, buf);
  float inv = 1.0f / ssum;
  probs[b * NK + tid] = e0 * inv;
  probs[b * NK + tid + 256] = e1 * inv;
}

// -------- halting, accumulate output, advance state -------------------------
__global__ void k_halt(const float* __restrict__ zf2, const float* __restrict__ Wh,
                       const float* __restrict__ bh, const float* __restrict__ zq,
                       float* __restrict__ acc, float* __restrict__ rem,
                       float* __restrict__ zf, int step) {
  __shared__ float buf[256];
  __shared__ float sh[2];
  int b = blockIdx.x, tid = threadIdx.x;
  float p = zf2[b * NF + tid] * Wh[tid] + zf2[b * NF + 256 + tid] * Wh[256 + tid];
  float dot = block_reduce_sum(p, buf);
  if (tid == 0) { sh[0] = sigm(dot + bh[0]); sh[1] = rem[b]; }
  __syncthreads();
  float ph = sh[0], r = sh[1];
  bool last = (step == NDEPTH - 1);
  float wgt = last ? r : ph * r;
  for (int j = tid; j < NF; j += 256) {
    float q = zq[b * NF + j];
    acc[b * NF + j] += wgt * q;
    zf[b * NF + j] = q;
  }
  if (tid == 0) rem[b] = last ? 0.0f : r * (1.0f - ph);
}

// ---------------------------------------------------------------------------
extern "C" void kernel_launch(void* const* d_in, const int* in_sizes, int n_in,
                              void* d_out, int out_size, void* d_ws, size_t ws_size,
                              hipStream_t stream) {
  const float* zr   = (const float*)d_in[0];
  const float* zi   = (const float*)d_in[1];
  const float* mem0 = (const float*)d_in[2];
  const float* ptr0 = (const float*)d_in[3];
  const float* Wq_r = (const float*)d_in[4];
  const float* Wq_i = (const float*)d_in[5];
  const float* Wk_r = (const float*)d_in[6];
  const float* Wk_i = (const float*)d_in[7];
  const float* Wv_r = (const float*)d_in[8];
  const float* Wv_i = (const float*)d_in[9];
  const float* Wc   = (const float*)d_in[10];
  const float* bc   = (const float*)d_in[11];
  const float* Wh   = (const float*)d_in[12];
  const float* bh   = (const float*)d_in[13];
  const float* cb   = (const float*)d_in[14];
  const float* adj  = (const float*)d_in[15];
  float* acc = (float*)d_out;               // (B, 2D) accumulator == output

  float* p = (float*)d_ws;                  // ~27 MB of fp32 scratch
  float* zf = p;      p += NB * NF;
  float* zf2 = p;     p += NB * NF;
  float* mem = p;     p += NB * NS * NF;
  float* ptr = p;     p += NB * NS;
  float* rem = p;     p += NB;
  float* probs = p;   p += NB * NK;
  float* mbar = p;    p += NB * NF;
  float* Rv = p;      p += NB * NF;
  float* G1 = p;      p += NB * NK;
  float* G2 = p;      p += NB * NK;
  float* zq = p;      p += NB * NF;
  float* magsum = p;  p += NB;
  float* magsq = p;   p += NB;
  float* znorm = p;   p += NB;
  float* scal = p;    p += 8;
  float* cbsq = p;    p += NK;
  float* Wq_pack = p; p += NF * NF;
  float* Wk_pack = p; p += NF * NF;
  float* Wv_pack = p; p += NF * NF;
  float* CbT_pack = p; p += NF * NK;
  float* Adj_pack = p; p += NK * NK;
  float* Cb_pack = p;  p += NK * NF;

  k_init<<<16384, 256, 0, stream>>>(zr, zi, mem0, ptr0, zf, mem, ptr, rem, probs, acc);
  k_pack_weff<<<512, 256, 0, stream>>>(Wq_r, Wq_i, Wq_pack);
  k_pack_weff<<<512, 256, 0, stream>>>(Wk_r, Wk_i, Wk_pack);
  k_pack_weff<<<512, 256, 0, stream>>>(Wv_r, Wv_i, Wv_pack);
  k_pack_mat<<<512, 256, 0, stream>>>(cb, CbT_pack, 1);   // B = codebook^T
  k_pack_mat<<<512, 256, 0, stream>>>(adj, Adj_pack, 0);  // B = adjacency
  k_pack_mat<<<512, 256, 0, stream>>>(cb, Cb_pack, 0);    // B = codebook
  k_cbsq<<<2, 256, 0, stream>>>(cb, cbsq);

  size_t attn_smem = (size_t)(3 * NS * NF + NS * NS + 2 * NS) * sizeof(float);
  dim3 ggemm(4, 8);   // M=256 -> 4 row blocks, N=512 -> 8 col blocks
  for (int t = 0; t < NDEPTH; t++) {
    k_ctrl<<<NB, 256, 0, stream>>>(Wc, bc, zf, ptr, mem);
    k_attn<<<NB, 256, attn_smem, stream>>>(mem, Wq_pack, Wk_pack, ptr, mbar);
    k_gemm<<<ggemm, 256, 0, stream>>>(mbar, Wv_pack, Rv);     // read = V(mbar)
    k_post<<<NB, 256, 0, stream>>>(zf, Rv, zf2, magsum, magsq, znorm);
    k_var<<<1, 256, 0, stream>>>(magsum, magsq, scal);
    k_gemm<<<ggemm, 256, 0, stream>>>(zf2, CbT_pack, G1);     // zf2 @ cb^T
    k_gemm<<<ggemm, 256, 0, stream>>>(probs, Adj_pack, G2);   // prev_probs @ adj
    k_dist<<<NB, 256, 0, stream>>>(G1, G2, cbsq, znorm, scal, probs, t);
    k_gemm<<<ggemm, 256, 0, stream>>>(probs, Cb_pack, zq);    // zq = probs @ cb
    k_halt<<<NB, 256, 0, stream>>>(zf2, Wh, bh, zq, acc, rem, zf, t);
  }
}